// SSDPostProcess_17051020165417
// MI455X (gfx1250) — compile-verified
//
#include <hip/hip_runtime.h>
#include <hip/hip_bf16.h>
#include <stdint.h>

// SSD post-process for MI455X (gfx1250, wave32).
// decode -> per-(b,c) top-512 select (scores read ONCE, cached in VGPRs;
// strided reads are L2-resident: 51MB < 192MB L2) -> per-lane greedy NMS
// (box+score tiles DMA'd to LDS by the Tensor Data Mover) -> per-batch
// top-100 merge (36.4KB score slab DMA'd to LDS by TDM).
#define BB   16
#define NN   8732
#define CC   91
#define KC   512
#define MD   100
#define NEGF (-1e30f)
#define S_TH 0.05f
#define I_TH 0.5f

typedef unsigned int u32;
typedef __attribute__((ext_vector_type(4))) u32 v4u;
typedef __attribute__((ext_vector_type(8))) int v8i;
typedef __attribute__((ext_vector_type(4))) int v4i;

// ---------------------------------------------------------------------------
// Tensor Data Mover: issue a DMA of `nbytes` (multiple of 4) of contiguous
// global memory into LDS at byte offset `lds_off`. D# per CDNA5 ISA §8:
//   group0: [1:0]=count=1, [63:32]=lds_addr, [120:64]=global_addr, type=2
//   group1: data_size=4B, tensor_dim0=tile_dim0=nbytes/4 (<=65535), tile_dim1=1,
//           tensor_dim0_stride=nbytes/4, workgroup_mask=0 (not in cluster)
// Tracked with TENSORcnt; caller waits with tdm_wait_all().
// ---------------------------------------------------------------------------
__device__ __forceinline__ void tdm_issue_load(uint64_t gaddr, u32 lds_off, u32 nbytes) {
  const u32 elems = nbytes >> 2;
  v4u g0 = { 1u,                                        // count=1 (valid)
             lds_off,                                   // lds_addr
             (u32)(gaddr & 0xffffffffu),                // global_addr lo
             (u32)((gaddr >> 32) & 0x01ffffffu) | (2u << 30) }; // ga hi | type=2
  v8i g1 = { (int)(2u << 16),                 // data_size=2 (4B); wg_mask=0
             (int)((elems & 0xffffu) << 16),  // tensor_dim0[15:0] @bits63:48
             (int)(1u << 16),                 // tensor_dim1=1 @bits95:80
             (int)((elems & 0xffffu) << 16),  // tile_dim0 @bits127:112
             1,                               // tile_dim1=1 @bits143:128
             (int)elems,                      // tensor_dim0_stride lo32
             0, 0 };
  v4i g2 = { 0, 0, 0, 0 };
  v4i g3 = { 0, 0, 0, 0 };
#if __clang_major__ >= 23
  v8i g4 = { 0, 0, 0, 0, 0, 0, 0, 0 };
  __builtin_amdgcn_tensor_load_to_lds(g0, g1, g2, g3, g4, 0);
#else
  __builtin_amdgcn_tensor_load_to_lds(g0, g1, g2, g3, 0);
#endif
}

__device__ __forceinline__ void tdm_wait_all() {
  __builtin_amdgcn_s_wait_tensorcnt(0);
}

__device__ __forceinline__ float sigmoidf_(float x) {
  return 1.0f / (1.0f + __expf(-x));
}

// ---------------------------------------------------------------------------
// Kernel 1: FasterRCNN box decode, one thread per (b,n). ~140K threads.
// ---------------------------------------------------------------------------
__global__ void decode_kernel(const float* __restrict__ rel,
                              const float* __restrict__ anch,
                              float* __restrict__ dec) {
  const int idx = blockIdx.x * blockDim.x + threadIdx.x;
  if (idx >= BB * NN) return;
  const int n = idx % NN;
  const float ay0 = anch[n * 4 + 0], ax0 = anch[n * 4 + 1];
  const float ay1 = anch[n * 4 + 2], ax1 = anch[n * 4 + 3];
  const float ha = ay1 - ay0, wa = ax1 - ax0;
  const float yca = ay0 + 0.5f * ha, xca = ax0 + 0.5f * wa;
  const float* r = rel + (size_t)idx * 4;
  const float ty = r[0] * 0.1f, tx = r[1] * 0.1f;
  const float th = r[2] * 0.2f, tw = r[3] * 0.2f;
  const float yc = ty * ha + yca, xc = tx * wa + xca;
  const float h = __expf(th) * ha, w = __expf(tw) * wa;
  float* d = dec + (size_t)idx * 4;
  d[0] = fminf(fmaxf(yc - 0.5f * h, 0.f), 320.f);
  d[1] = fminf(fmaxf(xc - 0.5f * w, 0.f), 320.f);
  d[2] = fminf(fmaxf(yc + 0.5f * h, 0.f), 320.f);
  d[3] = fminf(fmaxf(xc + 0.5f * w, 0.f), 320.f);
}

// ---------------------------------------------------------------------------
// Kernel 2: per-(b,c) top-512 selection. One 256-thread block (8 wave32) per
// lane, 1456 blocks. Scores are read from memory EXACTLY ONCE: each thread
// caches its 35 sigmoid values in VGPRs (fully unrolled -> register array).
// Radix-style: 256-bin histogram of the float's top byte (monotone for
// p in (0,1]), threshold bin, rank-stable compaction via block scan.
// ---------------------------------------------------------------------------
__global__ void select_topk_kernel(const float* __restrict__ scores,
                                   const float* __restrict__ dec,
                                   float* __restrict__ cand_s,
                                   float* __restrict__ cand_b) {
  constexpr int NPT = (NN + 255) / 256;   // 35 elements per thread
  const int lane = blockIdx.x;            // b*CC + c
  const int b = lane / CC, c = lane % CC;
  const int tid = threadIdx.x;
  const float* sp = scores + (size_t)b * NN * CC + c;

  __shared__ int hist[256];
  __shared__ int hiCnt[256];
  __shared__ int eqCnt[256];
  __shared__ int sThr, sHiTot;

  // Single pass over global memory; values kept in VGPRs.
  float pv[NPT];
#pragma unroll
  for (int i = 0; i < NPT; ++i) {
    const int n = tid + (i << 8);
    __builtin_prefetch(sp + (size_t)(n + 256) * CC, 0, 1);  // global_prefetch_b8
    pv[i] = (n < NN) ? sigmoidf_(sp[(size_t)n * CC]) : 0.0f;
  }

  hist[tid] = 0;
  __syncthreads();
#pragma unroll
  for (int i = 0; i < NPT; ++i) {
    const int n = tid + (i << 8);
    if (n < NN) atomicAdd(&hist[__float_as_uint(pv[i]) >> 24], 1);
  }
  __syncthreads();

  if (tid == 0) {
    int acc = 0, t = 0;
    for (int bin = 255; bin >= 0; --bin) {
      if (acc + hist[bin] >= KC) { t = bin; break; }   // NN >= KC: always hits
      acc += hist[bin];
    }
    sThr = t;
  }
  __syncthreads();
  const int t = sThr;

  // Per-thread counts of (bin > t) and (bin == t), from registers.
  int chi = 0, ceq = 0;
#pragma unroll
  for (int i = 0; i < NPT; ++i) {
    const int n = tid + (i << 8);
    if (n < NN) {
      const int bin = (int)(__float_as_uint(pv[i]) >> 24);
      chi += (bin > t);
      ceq += (bin == t);
    }
  }
  hiCnt[tid] = chi;
  eqCnt[tid] = ceq;
  __syncthreads();

  // Inclusive Hillis-Steele scan over 256 entries (deterministic ranks).
  for (int off = 1; off < 256; off <<= 1) {
    const int vh = (tid >= off) ? hiCnt[tid - off] : 0;
    const int ve = (tid >= off) ? eqCnt[tid - off] : 0;
    __syncthreads();
    hiCnt[tid] += vh;
    eqCnt[tid] += ve;
    __syncthreads();
  }
  const int baseHi = hiCnt[tid] - chi;
  const int baseEq = eqCnt[tid] - ceq;
  if (tid == 0) sHiTot = hiCnt[255];
  __syncthreads();
  const int hiTot = sHiTot;   // < KC by construction

  // Compaction: write candidate scores + gathered decoded boxes.
  float* cs = cand_s + (size_t)lane * KC;
  float* cb = cand_b + (size_t)lane * KC * 4;
  const float* db = dec + (size_t)b * NN * 4;
  int iHi = 0, iEq = 0;
#pragma unroll
  for (int i = 0; i < NPT; ++i) {
    const int n = tid + (i << 8);
    if (n < NN) {
      const float p = pv[i];
      const int bin = (int)(__float_as_uint(p) >> 24);
      int slot = -1;
      if (bin > t) {
        slot = baseHi + (iHi++);
      } else if (bin == t) {
        const int s2 = hiTot + baseEq + (iEq++);
        if (s2 < KC) slot = s2;
      }
      if (slot >= 0) {
        cs[slot] = p;
        cb[slot * 4 + 0] = db[n * 4 + 0];
        cb[slot * 4 + 1] = db[n * 4 + 1];
        cb[slot * 4 + 2] = db[n * 4 + 2];
        cb[slot * 4 + 3] = db[n * 4 + 3];
      }
    }
  }
}

// ---------------------------------------------------------------------------
// Kernel 3: greedy NMS per lane. Candidate box tile (8 KB) and score tile
// (2 KB) both DMA'd into LDS by the Tensor Data Mover (two descriptors,
// one TENSORcnt wait); then 100 iterations of LDS-tree argmax + IoU suppress.
// ---------------------------------------------------------------------------
struct NmsSmem {
  float bx[KC * 4];   // first member: TDM box tile targets LDS byte offset 0
  float sc[KC];       // TDM score tile targets offsetof(NmsSmem, sc)
  float rv[256];
  int   ri[256];
  float bbox[4];
  float bs;
  int   bj;
};

__global__ void nms_kernel(const float* __restrict__ cand_s,
                           const float* __restrict__ cand_b,
                           float* __restrict__ sel_s,
                           float* __restrict__ sel_b) {
  const int lane = blockIdx.x;
  const int tid = threadIdx.x;   // 256 threads
  __shared__ NmsSmem sm;

  // Wave 0 issues both DMAs back-to-back, then one wait covers both
  // (TENSORcnt is in-order within a wave).
  if (tid < warpSize) {
    tdm_issue_load((uint64_t)(uintptr_t)(cand_b + (size_t)lane * KC * 4),
                   0u, KC * 4 * sizeof(float));
    tdm_issue_load((uint64_t)(uintptr_t)(cand_s + (size_t)lane * KC),
                   (u32)__builtin_offsetof(NmsSmem, sc), KC * sizeof(float));
    tdm_wait_all();
  }
  __syncthreads();   // DMA'd tiles visible to all waves

  // Threshold scores in place.
  for (int k = tid; k < KC; k += 256) {
    const float p = sm.sc[k];
    sm.sc[k] = (p > S_TH) ? p : NEGF;
  }
  __syncthreads();

  for (int it = 0; it < MD; ++it) {
    // argmax over 512 (first-index tie-break, matching jnp.argmax)
    float v1 = sm.sc[tid];
    int i1 = tid;
    {
      const float v2 = sm.sc[tid + 256];
      if (v2 > v1) { v1 = v2; i1 = tid + 256; }
    }
    sm.rv[tid] = v1;
    sm.ri[tid] = i1;
    __syncthreads();
    for (int s = 128; s > 0; s >>= 1) {
      if (tid < s) {
        const float vo = sm.rv[tid + s];
        const int io = sm.ri[tid + s];
        if (vo > sm.rv[tid] || (vo == sm.rv[tid] && io < sm.ri[tid])) {
          sm.rv[tid] = vo;
          sm.ri[tid] = io;
        }
      }
      __syncthreads();
    }
    if (tid == 0) {
      const int j = sm.ri[0];
      const float s = sm.rv[0];
      sm.bj = j;
      sm.bs = s;
      sm.bbox[0] = sm.bx[j * 4 + 0];
      sm.bbox[1] = sm.bx[j * 4 + 1];
      sm.bbox[2] = sm.bx[j * 4 + 2];
      sm.bbox[3] = sm.bx[j * 4 + 3];
      sm.sc[j] = NEGF;
      const bool valid = s > NEGF * 0.5f;
      sel_s[(size_t)lane * MD + it] = valid ? s : NEGF;
      float* ob = sel_b + ((size_t)lane * MD + it) * 4;
      ob[0] = sm.bbox[0]; ob[1] = sm.bbox[1];
      ob[2] = sm.bbox[2]; ob[3] = sm.bbox[3];
    }
    __syncthreads();
    const bool valid = sm.bs > NEGF * 0.5f;
    if (valid) {
      const float by0 = sm.bbox[0], bx0 = sm.bbox[1];
      const float by1 = sm.bbox[2], bx1 = sm.bbox[3];
      const float area = (by1 - by0) * (bx1 - bx0);
#pragma unroll
      for (int half = 0; half < 2; ++half) {
        const int k = tid + half * 256;
        const float cy0 = sm.bx[k * 4 + 0], cx0 = sm.bx[k * 4 + 1];
        const float cy1 = sm.bx[k * 4 + 2], cx1 = sm.bx[k * 4 + 3];
        const float iy = fmaxf(fminf(by1, cy1) - fmaxf(by0, cy0), 0.f);
        const float ix = fmaxf(fminf(bx1, cx1) - fmaxf(bx0, cx0), 0.f);
        const float inter = iy * ix;
        const float areas = (cy1 - cy0) * (cx1 - cx0);
        const float iou = inter / (area + areas - inter + 1e-8f);
        if (iou > I_TH) sm.sc[k] = NEGF;
      }
    }
    __syncthreads();
  }
}

// ---------------------------------------------------------------------------
// Kernel 4: per-batch top-100 of the 91*100 selected scores. One block per
// batch; the 36.4 KB contiguous score slab is DMA'd to LDS by the TDM
// (tile_dim0 = 9100 elements < 2^16). Well inside the 320KB WGP LDS pool.
// ---------------------------------------------------------------------------
struct FinSmem {
  float s[CC * MD];   // first member: TDM slab targets LDS byte offset 0
  float rv[256];
  int   ri[256];
  int   nv;
};

__global__ void final_topk_kernel(const float* __restrict__ sel_s,
                                  const float* __restrict__ sel_b,
                                  float* __restrict__ out) {
  const int b = blockIdx.x;
  const int tid = threadIdx.x;   // 256 threads
  constexpr int M = CC * MD;     // 9100
  __shared__ FinSmem fm;

  if (tid < warpSize) {
    tdm_issue_load((uint64_t)(uintptr_t)(sel_s + (size_t)b * M),
                   0u, M * sizeof(float));
    tdm_wait_all();
  }
  if (tid == 0) fm.nv = 0;
  __syncthreads();

  float* out_b = out;                              // (16,100,4)
  float* out_s = out + (size_t)BB * MD * 4;        // (16,100)
  float* out_l = out_s + (size_t)BB * MD;          // (16,100)
  float* out_n = out_l + (size_t)BB * MD;          // (16,) written as float

  for (int it = 0; it < MD; ++it) {
    float v1 = -3.0e38f;
    int i1 = 0;
    for (int k = tid; k < M; k += 256) {
      const float vk = fm.s[k];
      if (vk > v1) { v1 = vk; i1 = k; }
    }
    fm.rv[tid] = v1;
    fm.ri[tid] = i1;
    __syncthreads();
    for (int st = 128; st > 0; st >>= 1) {
      if (tid < st) {
        const float vo = fm.rv[tid + st];
        const int io = fm.ri[tid + st];
        if (vo > fm.rv[tid] || (vo == fm.rv[tid] && io < fm.ri[tid])) {
          fm.rv[tid] = vo;
          fm.ri[tid] = io;
        }
      }
      __syncthreads();
    }
    if (tid == 0) {
      const int j = fm.ri[0];
      const float v = fm.rv[0];
      const bool valid = v > NEGF * 0.5f;
      out_s[(size_t)b * MD + it] = valid ? v : 0.f;
      out_l[(size_t)b * MD + it] = valid ? (float)(j / MD) : 0.f;
      const float* sb = sel_b + ((size_t)b * M + j) * 4;
      float* ob = out_b + ((size_t)b * MD + it) * 4;
      ob[0] = valid ? sb[0] : 0.f;
      ob[1] = valid ? sb[1] : 0.f;
      ob[2] = valid ? sb[2] : 0.f;
      ob[3] = valid ? sb[3] : 0.f;
      if (valid) fm.nv++;
      fm.s[j] = NEGF;
    }
    __syncthreads();
  }
  if (tid == 0) out_n[b] = (float)fm.nv;
}

// ---------------------------------------------------------------------------
extern "C" void kernel_launch(void* const* d_in, const int* in_sizes, int n_in,
                              void* d_out, int out_size, void* d_ws, size_t ws_size,
                              hipStream_t stream) {
  const float* rel    = (const float*)d_in[0];   // (16,8732,4)
  const float* scores = (const float*)d_in[1];   // (16,8732,91)
  const float* anch   = (const float*)d_in[2];   // (8732,4)

  // Workspace layout (floats): ~20.1 MB total
  float* ws = (float*)d_ws;
  size_t o = 0;
  float* dec    = ws + o; o += (size_t)BB * NN * 4;        //  558,848
  float* cand_s = ws + o; o += (size_t)BB * CC * KC;       //  745,472
  float* cand_b = ws + o; o += (size_t)BB * CC * KC * 4;   // 2,981,888
  float* sel_s  = ws + o; o += (size_t)BB * CC * MD;       //  145,600
  float* sel_b  = ws + o; o += (size_t)BB * CC * MD * 4;   //  582,400

  decode_kernel<<<(BB * NN + 255) / 256, 256, 0, stream>>>(rel, anch, dec);
  select_topk_kernel<<<BB * CC, 256, 0, stream>>>(scores, dec, cand_s, cand_b);
  nms_kernel<<<BB * CC, 256, 0, stream>>>(cand_s, cand_b, sel_s, sel_b);
  final_topk_kernel<<<BB, 256, 0, stream>>>(sel_s, sel_b, (float*)d_out);
}